// label_assign_56719338111715
// MI455X (gfx1250) — compile-verified
//
#include <hip/hip_runtime.h>
#include <cstdint>

// Problem constants (from reference setup_inputs)
#define L_ 3
#define B_ 16
#define M_ 64
#define N_ 9
#define W_ 64
#define H_ 64
#define C_ 20

#define TOTAL_ (L_ * B_ * M_)          // 3072 (lvl,b,m) items
#define ANCH_FLOATS_ (L_ * N_ * 4)     // 108 floats of anchor table

// Output section offsets (floats), concatenated in reference return order.
#define OFF_CLS_P_ 0
#define OFF_REG_P_ (TOTAL_ * C_)                       // 61440
#define OFF_CLS_T_ (TOTAL_ * C_ + TOTAL_ * 5)          // 76800
#define OFF_REG_T_ (TOTAL_ * (2 * C_ + 5))             // 138240

__global__ __launch_bounds__(128) void label_assign_56719338111715_kernel(
    const float* __restrict__ q_anchors,       // [L,N,4]
    const float* __restrict__ targets,         // [B,M,6]
    const float* __restrict__ regressions,     // [L,B,N,W,H,5]
    const float* __restrict__ classifications, // [L,B,N,W,H,20]
    float* __restrict__ out)
{
  __shared__ float sh_anchors[ANCH_FLOATS_];

  const int tid = threadIdx.x;

  // --- CDNA5 async global->LDS staging of the anchor table (ASYNCcnt path).
  // GV mode: global_load_async_to_lds_b32 vdst(lds byte addr), vaddr(64b), off
  if (tid < ANCH_FLOATS_) {
    // low 32 bits of a flat pointer into shared memory == LDS byte offset
    unsigned lds_off = (unsigned)(uintptr_t)(&sh_anchors[tid]);
    unsigned long long gaddr = (unsigned long long)(uintptr_t)(q_anchors + tid);
    asm volatile("global_load_async_to_lds_b32 %0, %1, off"
                 :: "v"(lds_off), "v"(gaddr)
                 : "memory");
  }

  const int i   = blockIdx.x * blockDim.x + tid;
  const int lvl = i / (B_ * M_);
  const int rem = i - lvl * (B_ * M_);
  const int b   = rem / M_;
  const int m   = rem - b * M_;

  const float* t = targets + (b * M_ + m) * 6;
  // Overlap the target-row fetch with the async LDS staging (global_prefetch_b8).
  __builtin_prefetch(t, 0, 0);

  // Wait for this wave's async transfers, then make staging visible block-wide.
  asm volatile("s_wait_asynccnt 0" ::: "memory");
  __syncthreads();

  if (i >= TOTAL_) return;

  // ---- target geometry
  const float x1 = t[0], y1 = t[1], x2 = t[2], y2 = t[3];
  const int   lab = (int)t[4];
  const float w = x2 - x1;
  const float h = y2 - y1;

  const float stride = (float)(32 >> lvl);      // 32, 16, 8
  const float hw = (0.5f * w) / stride;         // bbox = origin/stride (centered)
  const float hh = (0.5f * h) / stride;
  const int   gx = (int)hw;                     // grids = trunc(center/stride)
  const int   gy = (int)hh;
  const float area_b = (hw + hw) * (hh + hh);

  // ---- IoU argmax over the N=9 anchors of this level (anchors are centered)
  const float* anch = &sh_anchors[lvl * (N_ * 4)];
  int   arg  = 0;
  float best = -1.0f;
#pragma unroll
  for (int n = 0; n < N_; ++n) {
    const float a0 = anch[n * 4 + 0];
    const float a1 = anch[n * 4 + 1];
    const float a2 = anch[n * 4 + 2];
    const float a3 = anch[n * 4 + 3];
    const float ltx = fmaxf(a0, -hw);
    const float lty = fmaxf(a1, -hh);
    const float rbx = fminf(a2,  hw);
    const float rby = fminf(a3,  hh);
    const float iw  = fmaxf(rbx - ltx, 0.0f);
    const float ih  = fmaxf(rby - lty, 0.0f);
    const float inter  = iw * ih;
    const float area_a = (a2 - a0) * (a3 - a1);
    const float uni    = area_a + area_b - inter;
    const float iou    = inter / fmaxf(uni, 1e-6f);
    if (iou > best) { best = iou; arg = n; }   // strict '>' == first-max (jnp.argmax)
  }

  // ---- gather predictions at (b, arg, gx, gy)
  const int cell = (((lvl * B_ + b) * N_ + arg) * W_ + gx) * H_ + gy;
  const float* cp = classifications + (long long)cell * C_;   // 80B-aligned row
  const float* rp = regressions    + (long long)cell * 5;

  const int om = (lvl * B_ + b) * M_ + m;   // row index in [0, TOTAL_)
  float* out_cls_p = out + OFF_CLS_P_ + (long long)om * C_;   // 80B-aligned
  float* out_reg_p = out + OFF_REG_P_ + (long long)om * 5;
  float* out_cls_t = out + OFF_CLS_T_ + (long long)om * C_;   // 80B-aligned
  float* out_reg_t = out + OFF_REG_T_ + (long long)om * 4;

  // cls_p: 20 floats = 5 x float4 (row byte offset is a multiple of 80 -> 16B aligned)
  {
    const float4* cp4 = (const float4*)cp;
    float4* op4 = (float4*)out_cls_p;
#pragma unroll
    for (int k = 0; k < 5; ++k) op4[k] = cp4[k];
  }

  // reg_p: 5 floats (only 4B alignment guaranteed)
#pragma unroll
  for (int k = 0; k < 5; ++k) out_reg_p[k] = rp[k];

  // cls_t: one-hot of 20, written as 5 x float4
  {
    float4* ot4 = (float4*)out_cls_t;
#pragma unroll
    for (int k = 0; k < 5; ++k) {
      float4 v;
      v.x = (4 * k + 0 == lab) ? 1.0f : 0.0f;
      v.y = (4 * k + 1 == lab) ? 1.0f : 0.0f;
      v.z = (4 * k + 2 == lab) ? 1.0f : 0.0f;
      v.w = (4 * k + 3 == lab) ? 1.0f : 0.0f;
      ot4[k] = v;
    }
  }

  // reg_t: encode(assign, target)
  {
    const float aw  = anch[arg * 4 + 2] - anch[arg * 4 + 0];
    const float ah  = anch[arg * 4 + 3] - anch[arg * 4 + 1];
    const float exw = fmaxf(aw, 1.0f);
    const float exh = fmaxf(ah, 1.0f);
    const float gtw = fmaxf(w, 1.0f);
    const float gth = fmaxf(h, 1.0f);
    const float gcx = x1 + 0.5f * gtw;
    const float gcy = y1 + 0.5f * gth;
    // fractional part via truncating int16 cast (matches astype(int16) round-trip)
    const float dx = gcx - (float)(short)gcx;
    const float dy = gcy - (float)(short)gcy;
    out_reg_t[0] = dx;
    out_reg_t[1] = dy;
    out_reg_t[2] = logf(gtw / exw);
    out_reg_t[3] = logf(gth / exh);
  }
}

extern "C" void kernel_launch(void* const* d_in, const int* in_sizes, int n_in,
                              void* d_out, int out_size, void* d_ws, size_t ws_size,
                              hipStream_t stream) {
  (void)in_sizes; (void)n_in; (void)out_size; (void)d_ws; (void)ws_size;
  const float* q_anchors       = (const float*)d_in[0];
  const float* targets         = (const float*)d_in[1];
  const float* regressions     = (const float*)d_in[2];
  const float* classifications = (const float*)d_in[3];
  float* out = (float*)d_out;

  const int block = 128;                         // 4 wave32 waves
  const int grid  = (TOTAL_ + block - 1) / block; // 24 blocks
  label_assign_56719338111715_kernel<<<grid, block, 0, stream>>>(
      q_anchors, targets, regressions, classifications, out);
}